// GatedGCN_53180285059709
// MI455X (gfx1250) — compile-verified
//
#include <hip/hip_runtime.h>
#include <hip/hip_bf16.h>

#define NN 100000
#define NE 1000000
#define D 64
#define LN_EPS 1e-5f
#define DIV_EPS 1e-6f

typedef float v2f __attribute__((ext_vector_type(2)));
typedef float v8f __attribute__((ext_vector_type(8)));

// D = A(16x4, f32) * B(4x16, f32) + C(16x16, f32)
static __device__ __forceinline__ v8f wmma_k4(v2f a, v2f b, v8f c) {
    return __builtin_amdgcn_wmma_f32_16x16x4_f32(false, a, false, b, (short)0, c, false, false);
}

// LDS byte offset of a __shared__ pointer. On AMDGPU the low 32 bits of the
// flat LDS-aperture address are the hardware LDS byte address (ISA 10.2).
// Crucially, the ptrtoint makes the LDS object ESCAPE into the asm operand,
// so the compiler must assume the async-load asm stores into it (otherwise a
// never-stored LDS array folds its loads to undef -- observed in round 3).
static __device__ __forceinline__ unsigned lds_off(const void* p) {
    return (unsigned)(uintptr_t)p;
}

// Async-stage 16 KB (one 64x64 f32 matrix) into LDS at byte address ldsBase.
static __device__ __forceinline__ void stage_async(const float* __restrict__ W,
                                                   unsigned ldsBase, unsigned tid) {
#pragma unroll
    for (int it = 0; it < 4; ++it) {
        const unsigned idx16 = it * 256u + tid;                  // 0..1023 16B chunks
        const unsigned ldsoff = ldsBase + idx16 * 16u;
        const unsigned goff = idx16 * 16u;
        asm volatile("global_load_async_to_lds_b128 %0, %1, %2"
                     :: "v"(ldsoff), "v"(goff), "s"(W)
                     : "memory");
    }
}
static __device__ __forceinline__ void wait_async() {
    asm volatile("s_wait_asynccnt 0" ::: "memory");
}

// ---------------------------------------------------------------------------
// Kernel 0: zero the segment-sum accumulators (ws is poisoned by harness)
// ---------------------------------------------------------------------------
__global__ __launch_bounds__(256) void zero_kernel(float4* __restrict__ p, int n4) {
    int i = blockIdx.x * blockDim.x + threadIdx.x;
    if (i < n4) p[i] = make_float4(0.f, 0.f, 0.f, 0.f);
}

// ---------------------------------------------------------------------------
// Kernel 1: fused node projections  A1h,A2h,B1h,B2h = h @ {A1w,A2w,B1w,B2w} + b
// One wave = one 16-row tile of h (h read once for all 4 matrices).
// Weights double-buffered in LDS via async loads: stage(w+1) overlaps
// compute(w). Matrix loop fully unrolled -> pointers bind to kernel args.
// ---------------------------------------------------------------------------
__global__ __launch_bounds__(256) void node_proj_kernel(
    const float* __restrict__ h,
    const float* __restrict__ A1w, const float* __restrict__ A1b,
    const float* __restrict__ A2w, const float* __restrict__ A2b,
    const float* __restrict__ B1w, const float* __restrict__ B1b,
    const float* __restrict__ B2w, const float* __restrict__ B2b,
    float* __restrict__ A1h, float* __restrict__ A2h,
    float* __restrict__ B1h, float* __restrict__ B2h)
{
    __shared__ float sW[2][D * D];                               // 2 x 16 KB
    const int nTiles = NN / 16;                                  // 6250
    const int wave = blockIdx.x * (blockDim.x >> 5) + (threadIdx.x >> 5);
    const bool active = wave < nTiles;                           // wave-uniform
    const int lane = threadIdx.x & 31;
    const int half = lane >> 4;
    const int mrow = lane & 15;
    const int row0 = wave * 16;
    const unsigned base0 = lds_off(&sW[0][0]);
    const unsigned base1 = lds_off(&sW[1][0]);

    // Kick off stage of matrix 0 into buffer 0 immediately.
    stage_async(A1w, base0, threadIdx.x);

    // Overlap: A fragments. 16 frags of 16x4; lanes 0-15 hold (K=4f+0,4f+1),
    // lanes 16-31 hold (K=4f+2,4f+3) of row M=lane&15.
    v2f a[16];
    if (active) {
        const float* hrow = h + (size_t)(row0 + mrow) * D;
        __builtin_prefetch(hrow + (size_t)128 * D, 0, 0);
#pragma unroll
        for (int f = 0; f < 16; ++f) {
            const int k = 4 * f + 2 * half;
            a[f].x = hrow[k];
            a[f].y = hrow[k + 1];
        }
    }

    wait_async();
    __syncthreads();                                             // buf0 ready

    const float* Ws[4] = {A1w, A2w, B1w, B2w};
    const float* Bs[4] = {A1b, A2b, B1b, B2b};
    float*       Os[4] = {A1h, A2h, B1h, B2h};

#pragma unroll
    for (int w = 0; w < 4; ++w) {
        const int cur = w & 1;
        // Prefetch next matrix into the other buffer (safe: barrier at end of
        // previous iteration guarantees no one still reads that buffer).
        if (w < 3) stage_async(Ws[w + 1], cur ? base0 : base1, threadIdx.x);

        if (active) {
            const float* bias = Bs[w];
            float* Out = Os[w];
            const float* lw = &sW[cur][0];
#pragma unroll
            for (int nt = 0; nt < 4; ++nt) {
                const int col = nt * 16 + mrow;                  // N = lane&15 (both halves)
                v2f b[16];                                       // batch B frags from LDS
#pragma unroll
                for (int f = 0; f < 16; ++f) {
                    const int k = 4 * f + 2 * half;              // B: K = 2*half + vgpr
                    b[f].x = lw[k * D + col];
                    b[f].y = lw[(k + 1) * D + col];
                }
                const float bv = bias[col];
                v8f c;
#pragma unroll
                for (int i = 0; i < 8; ++i) c[i] = bv;           // bias uniform over M
#pragma unroll
                for (int f = 0; f < 16; ++f) c = wmma_k4(a[f], b[f], c);
                // C layout: VGPR i -> M = i + 8*half, N = lane&15
#pragma unroll
                for (int i = 0; i < 8; ++i)
                    Out[(size_t)(row0 + i + 8 * half) * D + col] = c[i];
            }
        }
        wait_async();                                            // next buf staged
        __syncthreads();
    }
}

// ---------------------------------------------------------------------------
// Kernel 2: fused edge pass.
//   B3e = e @ B3w + b3 (WMMA, 16 edges per wave; B3w async-staged in LDS)
//   e_ji = relu(LN(B1h[src]+B2h[dst]+B3e)) + e   -> write
//   sigma = sigmoid(e_ji); atomic segment sums at dst.
// ---------------------------------------------------------------------------
__global__ __launch_bounds__(256) void edge_kernel(
    const float* __restrict__ e,
    const int* __restrict__ src, const int* __restrict__ dst,
    const float* __restrict__ B3w, const float* __restrict__ B3b,
    const float* __restrict__ B1h, const float* __restrict__ B2h,
    const float* __restrict__ A2h,
    const float* __restrict__ ln_g, const float* __restrict__ ln_b,
    float* __restrict__ e_out,
    float* __restrict__ ssh, float* __restrict__ ss)
{
    __shared__ float sW[D * D];                                  // 16 KB
    const int nTiles = NE / 16;                                  // 62500
    const int wave = blockIdx.x * (blockDim.x >> 5) + (threadIdx.x >> 5);
    const bool active = wave < nTiles;                           // wave-uniform
    const int lane = threadIdx.x & 31;
    const int half = lane >> 4;
    const int mrow = lane & 15;
    const int row0 = wave * 16;

    stage_async(B3w, lds_off(sW), threadIdx.x);                  // async weight stage

    // Overlap: A fragments from 16 consecutive e rows while async runs.
    v2f a[16];
    if (active) {
        const float* erow = e + (size_t)(row0 + mrow) * D;
        __builtin_prefetch(erow + (size_t)128 * D, 0, 0);
#pragma unroll
        for (int f = 0; f < 16; ++f) {
            const int k = 4 * f + 2 * half;
            a[f].x = erow[k];
            a[f].y = erow[k + 1];
        }
    }

    wait_async();
    __syncthreads();
    if (!active) return;                                         // no barriers below

    // B3e tile: 4 accumulators live (full 64-wide rows needed for LN)
    v8f acc[4];
#pragma unroll
    for (int nt = 0; nt < 4; ++nt) {
        const int col = nt * 16 + mrow;
        v2f b[16];
#pragma unroll
        for (int f = 0; f < 16; ++f) {
            const int k = 4 * f + 2 * half;
            b[f].x = sW[k * D + col];
            b[f].y = sW[(k + 1) * D + col];
        }
        const float bv = B3b[col];
#pragma unroll
        for (int i = 0; i < 8; ++i) acc[nt][i] = bv;
#pragma unroll
        for (int f = 0; f < 16; ++f) acc[nt] = wmma_k4(a[f], b[f], acc[nt]);
    }

    // Epilogue: each v -> two edge rows (one per 16-lane half).
    // C layout: VGPR v holds row M=v+8*half, col = nt*16 + (lane&15).
#pragma unroll
    for (int v = 0; v < 8; ++v) {
        const int g = row0 + v + 8 * half;
        const int s = src[g];
        const int d = dst[g];
        const size_t sBase = (size_t)s * D;
        const size_t dBase = (size_t)d * D;
        const size_t gBase = (size_t)g * D;

        float x[4];
        float part = 0.f;
#pragma unroll
        for (int nt = 0; nt < 4; ++nt) {
            const int col = nt * 16 + mrow;
            const float val = acc[nt][v] + B1h[sBase + col] + B2h[dBase + col];
            x[nt] = val;
            part += val;
        }
        // LN over 64 cols: 4 local + 16-lane reduction (xor<16 stays in half)
        float mu = part;
#pragma unroll
        for (int o = 1; o < 16; o <<= 1) mu += __shfl_xor(mu, o, 32);
        mu *= (1.0f / 64.0f);

        float vp = 0.f;
#pragma unroll
        for (int nt = 0; nt < 4; ++nt) {
            const float dx = x[nt] - mu;
            vp += dx * dx;
        }
#pragma unroll
        for (int o = 1; o < 16; o <<= 1) vp += __shfl_xor(vp, o, 32);
        const float rstd = __frsqrt_rn(vp * (1.0f / 64.0f) + LN_EPS);

#pragma unroll
        for (int nt = 0; nt < 4; ++nt) {
            const int col = nt * 16 + mrow;
            const float ln = (x[nt] - mu) * rstd * ln_g[col] + ln_b[col];
            const float eo = fmaxf(ln, 0.0f) + e[gBase + col];   // relu + residual
            e_out[gBase + col] = eo;
            const float sg = 1.0f / (1.0f + __expf(-eo));        // sigmoid(e_ji)
            atomicAdd(&ss[dBase + col], sg);
            atomicAdd(&ssh[dBase + col], A2h[sBase + col] * sg);
        }
    }
}

// ---------------------------------------------------------------------------
// Kernel 3: node update. One wave per node, 2 cols/lane, full-wave LN.
// ---------------------------------------------------------------------------
__global__ __launch_bounds__(256) void node_update_kernel(
    const float* __restrict__ h,
    const float* __restrict__ A1h,
    const float* __restrict__ ssh, const float* __restrict__ ss,
    const float* __restrict__ ln_g, const float* __restrict__ ln_b,
    float* __restrict__ h_out)
{
    const int node = blockIdx.x * (blockDim.x >> 5) + (threadIdx.x >> 5);
    if (node >= NN) return;
    const int lane = threadIdx.x & 31;
    const size_t base = (size_t)node * D + 2 * lane;

    const float x0 = A1h[base]     + ssh[base]     / (ss[base]     + DIV_EPS);
    const float x1 = A1h[base + 1] + ssh[base + 1] / (ss[base + 1] + DIV_EPS);

    float mu = x0 + x1;
#pragma unroll
    for (int o = 1; o < 32; o <<= 1) mu += __shfl_xor(mu, o, 32);
    mu *= (1.0f / 64.0f);

    const float d0 = x0 - mu, d1 = x1 - mu;
    float vp = d0 * d0 + d1 * d1;
#pragma unroll
    for (int o = 1; o < 32; o <<= 1) vp += __shfl_xor(vp, o, 32);
    const float rstd = __frsqrt_rn(vp * (1.0f / 64.0f) + LN_EPS);

    const int c0 = 2 * lane, c1 = 2 * lane + 1;
    h_out[base]     = fmaxf(d0 * rstd * ln_g[c0] + ln_b[c0], 0.0f) + h[base];
    h_out[base + 1] = fmaxf(d1 * rstd * ln_g[c1] + ln_b[c1], 0.0f) + h[base + 1];
}

// ---------------------------------------------------------------------------
extern "C" void kernel_launch(void* const* d_in, const int* in_sizes, int n_in,
                              void* d_out, int out_size, void* d_ws, size_t ws_size,
                              hipStream_t stream) {
    (void)in_sizes; (void)n_in; (void)out_size; (void)ws_size;
    const float* h    = (const float*)d_in[0];
    const float* e    = (const float*)d_in[1];
    const int*   src  = (const int*)d_in[2];
    const int*   dst  = (const int*)d_in[3];
    const float* A1w  = (const float*)d_in[4];
    const float* A1b  = (const float*)d_in[5];
    const float* A2w  = (const float*)d_in[6];
    const float* A2b  = (const float*)d_in[7];
    const float* B1w  = (const float*)d_in[8];
    const float* B1b  = (const float*)d_in[9];
    const float* B2w  = (const float*)d_in[10];
    const float* B2b  = (const float*)d_in[11];
    const float* B3w  = (const float*)d_in[12];
    const float* B3b  = (const float*)d_in[13];
    const float* lneg = (const float*)d_in[14];
    const float* lneb = (const float*)d_in[15];
    const float* lnhg = (const float*)d_in[16];
    const float* lnhb = (const float*)d_in[17];

    const size_t ND = (size_t)NN * D;
    float* ws  = (float*)d_ws;
    float* A1h = ws;
    float* A2h = ws + ND;
    float* B1h = ws + 2 * ND;
    float* B2h = ws + 3 * ND;
    float* ssh = ws + 4 * ND;
    float* ssg = ws + 5 * ND;

    float* h_out = (float*)d_out;
    float* e_out = (float*)d_out + ND;

    // 1) zero accumulators (2*ND floats = 3,200,000 float4)
    {
        const int n4 = (int)(2 * ND / 4);
        zero_kernel<<<(n4 + 255) / 256, 256, 0, stream>>>((float4*)ssh, n4);
    }
    // 2) node projections: 6250 tiles, 8 waves/block
    node_proj_kernel<<<(NN / 16 + 7) / 8, 256, 0, stream>>>(
        h, A1w, A1b, A2w, A2b, B1w, B1b, B2w, B2b, A1h, A2h, B1h, B2h);
    // 3) fused edge pass: 62500 tiles, 8 waves/block
    edge_kernel<<<(NE / 16 + 7) / 8, 256, 0, stream>>>(
        e, src, dst, B3w, B3b, B1h, B2h, A2h, lneg, lneb, e_out, ssh, ssg);
    // 4) node update: 100000 nodes, 8 waves/block
    node_update_kernel<<<(NN + 7) / 8, 256, 0, stream>>>(
        h, A1h, ssh, ssg, lnhg, lnhb, h_out);
}